// PennyLaneQuantumLayer_22711787062008
// MI455X (gfx1250) — compile-verified
//
#include <hip/hip_runtime.h>
#include <hip/hip_bf16.h>

typedef __attribute__((ext_vector_type(2))) float v2f;
typedef __attribute__((ext_vector_type(8))) float v8f;

#define NQ      12
#define DIM     4096          // 2^12 amplitudes
#define NLAYERS 6
#define BLOCK   256           // 8 waves (wave32)
#define BATCH   2048

// Inverse of the all-pairs-CNOT permutation: Gray decode (suffix XOR).
__device__ __forceinline__ int gray_decode(int y) {
    int t = y;
    t ^= t >> 1; t ^= t >> 2; t ^= t >> 4; t ^= t >> 8;
    return t & (DIM - 1);
}

// Column c in [0,256) -> flat base address; contracted nibble m adds m*stride.
// g=0: bits 11..8 (wires 0..3), g=1: bits 7..4, g=2: bits 3..0.
__device__ __forceinline__ int group_base(int g, int c) {
    if (g == 0) return c;                              // addr = m*256 + c
    if (g == 1) return ((c >> 4) << 8) | (c & 15);     // addr = hi*256 + m*16 + lo
    return c << 4;                                     // addr = c*16 + m
}
__device__ __forceinline__ int group_stride(int g) {
    return (g == 0) ? 256 : ((g == 1) ? 16 : 1);
}

__global__ __launch_bounds__(BLOCK)
void qsim_kernel(const float* __restrict__ x,      // (BATCH, 12)
                 const float* __restrict__ wts,    // (6, 12, 3)
                 float* __restrict__ out)          // (BATCH, 2)
{
    extern __shared__ float smem[];
    float* S0R = smem;                 // 4096 state re (buffer A)
    float* S0I = S0R + DIM;            // 4096 state im
    float* S1R = S0I + DIM;            // 4096 (buffer B, for permutation)
    float* S1I = S1R + DIM;
    float* UR  = S1I + DIM;            // 3 * 256  fused 16x16 unitaries (re)
    float* UI  = UR + 3 * 256;         // 3 * 256  (im)
    float* GR  = UI + 3 * 256;         // 12 gates * 4 entries (re)
    float* GI  = GR + 48;              // 12 gates * 4 entries (im)

    const int tid  = threadIdx.x;
    const int b    = blockIdx.x;
    const int lane = tid & 31;
    const int wv   = tid >> 5;         // wave id 0..7
    const int nn   = lane & 15;        // col/row within tile
    const int hh   = lane >> 4;        // K-half selector

    float* curR = S0R; float* curI = S0I;
    float* altR = S1R; float* altI = S1I;

    // |00..0>
    for (int k = tid; k < DIM; k += BLOCK) { curR[k] = 0.0f; curI[k] = 0.0f; }
    if (tid == 0) curR[0] = 1.0f;

    // stage 0: RX angle-encoding; stages 1..6: StronglyEntangling layers
    for (int stage = 0; stage < 1 + NLAYERS; ++stage) {
        __syncthreads();

        // ---- build the 12 one-qubit gate matrices for this stage ----
        if (tid < NQ) {
            float m00r, m00i, m01r, m01i, m10r, m10i, m11r, m11i;
            if (stage == 0) {
                // RX(t): [[c, -i s], [-i s, c]]
                float t = 0.5f * x[b * NQ + tid];
                float c = __cosf(t), s = __sinf(t);
                m00r = c;    m00i = 0.0f;
                m01r = 0.0f; m01i = -s;
                m10r = 0.0f; m10i = -s;
                m11r = c;    m11i = 0.0f;
            } else {
                // Rot(phi,theta,omega) = RZ(omega) RY(theta) RZ(phi)
                const float* wp = wts + ((stage - 1) * NQ + tid) * 3;
                float phi = wp[0], th = wp[1], om = wp[2];
                float c = __cosf(0.5f * th), s = __sinf(0.5f * th);
                float a = 0.5f * (phi + om), bb = 0.5f * (phi - om);
                float ca = __cosf(a), sa = __sinf(a);
                float cb = __cosf(bb), sb = __sinf(bb);
                m00r =  c * ca;  m00i = -c * sa;   //  e^{-ia} c
                m01r = -s * cb;  m01i = -s * sb;   // -e^{+ib} s
                m10r =  s * cb;  m10i = -s * sb;   //  e^{-ib} s
                m11r =  c * ca;  m11i =  c * sa;   //  e^{+ia} c
            }
            GR[tid * 4 + 0] = m00r; GI[tid * 4 + 0] = m00i;
            GR[tid * 4 + 1] = m01r; GI[tid * 4 + 1] = m01i;
            GR[tid * 4 + 2] = m10r; GI[tid * 4 + 2] = m10i;
            GR[tid * 4 + 3] = m11r; GI[tid * 4 + 3] = m11i;
        }
        __syncthreads();

        // ---- fuse 4 gates per nibble into three 16x16 complex unitaries ----
        {
            int r = tid >> 4, sv = tid & 15;
            #pragma unroll
            for (int g = 0; g < 3; ++g) {
                float pr = 1.0f, pi = 0.0f;
                #pragma unroll
                for (int j = 0; j < 4; ++j) {           // j=0 is nibble MSB -> wire 4g+j
                    int wq = 4 * g + j;
                    int bi = (r  >> (3 - j)) & 1;
                    int bj = (sv >> (3 - j)) & 1;
                    float gr = GR[wq * 4 + bi * 2 + bj];
                    float gi = GI[wq * 4 + bi * 2 + bj];
                    float nr = pr * gr - pi * gi;
                    float ni = pr * gi + pi * gr;
                    pr = nr; pi = ni;
                }
                UR[g * 256 + tid] = pr;                  // U[r][sv], row-major
                UI[g * 256 + tid] = pi;
            }
        }
        __syncthreads();

        // ---- apply U_g along each nibble axis: batched complex 16x16 matmuls ----
        #pragma unroll
        for (int g = 0; g < 3; ++g) {
            const int str = group_stride(g);
            #pragma unroll
            for (int tt = 0; tt < 2; ++tt) {            // 16 tiles / 8 waves
                const int tile = wv * 2 + tt;
                const int c    = tile * 16 + nn;        // this lane's column
                const int base = group_base(g, c);
                v8f accR = {}; v8f accI = {};
                #pragma unroll
                for (int kk = 0; kk < 4; ++kk) {        // K accumulation, K=4 each
                    const int k0 = kk * 4 + 2 * hh;
                    v2f aR  = { UR[g * 256 + nn * 16 + k0], UR[g * 256 + nn * 16 + k0 + 1] };
                    v2f aI  = { UI[g * 256 + nn * 16 + k0], UI[g * 256 + nn * 16 + k0 + 1] };
                    v2f aIn = { -aI.x, -aI.y };          // f32 WMMA has no A-neg modifier
                    const int ad0 = base + k0 * str;
                    const int ad1 = ad0 + str;
                    v2f bR = { curR[ad0], curR[ad1] };
                    v2f bI = { curI[ad0], curI[ad1] };
                    // D_r += Ar*Br - Ai*Bi ; D_i += Ar*Bi + Ai*Br
                    accR = __builtin_amdgcn_wmma_f32_16x16x4_f32(false, aR,  false, bR, (short)0, accR, false, false);
                    accR = __builtin_amdgcn_wmma_f32_16x16x4_f32(false, aIn, false, bI, (short)0, accR, false, false);
                    accI = __builtin_amdgcn_wmma_f32_16x16x4_f32(false, aR,  false, bI, (short)0, accI, false, false);
                    accI = __builtin_amdgcn_wmma_f32_16x16x4_f32(false, aI,  false, bR, (short)0, accI, false, false);
                }
                // write back in place (tile's reads all precede its writes; tiles disjoint)
                #pragma unroll
                for (int v = 0; v < 8; ++v) {
                    const int r  = v + 8 * hh;
                    const int ad = base + r * str;
                    curR[ad] = accR[v];
                    curI[ad] = accI[v];
                }
            }
            __syncthreads();                            // nibble axes must serialize
        }

        // ---- all-pairs CNOT entangler == Gray-code permutation (layers only) ----
        if (stage > 0) {
            for (int y = tid; y < DIM; y += BLOCK) {
                int d = gray_decode(y);
                altR[y] = curR[d];
                altI[y] = curI[d];
            }
            __syncthreads();
            float* t0 = curR; curR = altR; altR = t0;
            float* t1 = curI; curI = altI; altI = t1;
        }
    }

    // ---- <Z> on wires 0 (bit 11) and 1 (bit 10) ----
    float p0 = 0.0f, p1 = 0.0f;
    for (int k = tid; k < DIM; k += BLOCK) {
        float pr = curR[k], pi = curI[k];
        float pp = pr * pr + pi * pi;
        p0 += ((k >> 11) & 1) ? -pp : pp;
        p1 += ((k >> 10) & 1) ? -pp : pp;
    }
    altR[tid] = p0; altI[tid] = p1;
    __syncthreads();
    #pragma unroll
    for (int s = 128; s > 0; s >>= 1) {
        if (tid < s) { altR[tid] += altR[tid + s]; altI[tid] += altI[tid + s]; }
        __syncthreads();
    }
    if (tid == 0) {
        out[b * 2 + 0] = altR[0];
        out[b * 2 + 1] = altI[0];
    }
}

extern "C" void kernel_launch(void* const* d_in, const int* in_sizes, int n_in,
                              void* d_out, int out_size, void* d_ws, size_t ws_size,
                              hipStream_t stream) {
    (void)in_sizes; (void)n_in; (void)d_ws; (void)ws_size; (void)out_size;
    const float* x   = (const float*)d_in[0];   // (2048, 12) float32
    const float* wts = (const float*)d_in[1];   // (6, 12, 3) float32
    float*       out = (float*)d_out;           // (2048, 2)  float32

    // dynamic LDS: 2x(re+im) state + 3 fused 16x16 unitaries (re+im) + 12 gates (re+im)
    const size_t shmem = (size_t)(4 * DIM + 2 * 3 * 256 + 2 * 48) * sizeof(float);
    qsim_kernel<<<BATCH, BLOCK, shmem, stream>>>(x, wts, out);
}